// NoduleRecallLoss_70033736728636
// MI455X (gfx1250) — compile-verified
//
#include <hip/hip_runtime.h>
#include <cstdint>

#define SDIM 192
#define S3   (192LL * 192LL * 192LL)   /* 7,077,888 voxels per (b,ch) volume */
#define MAXC 256                       /* accumulator slots (ncomp = 128)     */
#define BLOCK 256
#define VPT   8                        /* voxels per thread per tile          */
#define TILE_VOX (BLOCK * VPT)         /* 2048 voxels = 8KB of labels / tile  */

// ---------------------------------------------------------------------------
// Kernel 1: zero the global accumulators in workspace (ws is poisoned 0xAA).
// ---------------------------------------------------------------------------
__global__ void NoduleRecall_zero_ws(float* __restrict__ tp,
                                     unsigned* __restrict__ cnt) {
    int t = threadIdx.x;
    if (t < MAXC) { tp[t] = 0.0f; cnt[t] = 0u; }
}

// ---------------------------------------------------------------------------
// Kernel 2: stream labels via async global->LDS (double buffered, per-wave
// ASYNCcnt pipelining, non-temporal policy on the single-pass stream),
// conditionally fetch fg only under non-zero labels, accumulate per-component
// sums/counts with LDS atomics, flush once per block to global atomics.
// ---------------------------------------------------------------------------
__device__ __forceinline__ void async_tile_load(int* lds_dst,
                                                const int* gsrc) {
    // 32 bytes per lane = two b128 async loads per wave (ASYNCcnt += 2).
    unsigned l0 = (unsigned)(uintptr_t)lds_dst;
    unsigned l1 = l0 + 16u;
    unsigned long long g0 = (unsigned long long)(const void*)gsrc;
    unsigned long long g1 = g0 + 16ull;
    asm volatile("global_load_async_to_lds_b128 %0, %1, off th:TH_LOAD_NT"
                 :: "v"(l0), "v"(g0) : "memory");
    asm volatile("global_load_async_to_lds_b128 %0, %1, off th:TH_LOAD_NT"
                 :: "v"(l1), "v"(g1) : "memory");
}

__global__ __launch_bounds__(BLOCK)
void NoduleRecall_accum(const int* __restrict__ labels,
                        const float* __restrict__ x,
                        float* __restrict__ g_tp,
                        unsigned* __restrict__ g_cnt,
                        long long ntiles) {
    alignas(16) __shared__ int lbuf[2][TILE_VOX];    // 2 x 8KB label tiles
    __shared__ float    s_tp[MAXC];
    __shared__ unsigned s_cnt[MAXC];

    const int tid = threadIdx.x;
    s_tp[tid]  = 0.0f;
    s_cnt[tid] = 0u;
    __syncthreads();

    long long tile = blockIdx.x;

    // Prologue: issue async loads of first tile.
    if (tile < ntiles)
        async_tile_load(&lbuf[0][tid * VPT], labels + tile * TILE_VOX + tid * VPT);

    int buf = 0;
    for (; tile < ntiles; tile += gridDim.x) {
        long long next = tile + (long long)gridDim.x;
        if (next < ntiles) {
            async_tile_load(&lbuf[buf ^ 1][tid * VPT],
                            labels + next * TILE_VOX + tid * VPT);
            // 2 outstanding = next tile's pair; current tile is resident.
            asm volatile("s_wait_asynccnt 0x2" ::: "memory");
        } else {
            asm volatile("s_wait_asynccnt 0x0" ::: "memory");
        }

        // Each lane consumes exactly the 32B it loaded itself (no barrier
        // needed: ASYNCcnt is per-wave and async loads complete in order).
        const int4 lab0 = *reinterpret_cast<const int4*>(&lbuf[buf][tid * VPT]);
        const int4 lab1 = *reinterpret_cast<const int4*>(&lbuf[buf][tid * VPT + 4]);

        if (lab0.x | lab0.y | lab0.z | lab0.w | lab1.x | lab1.y | lab1.z | lab1.w) {
            // Tiles never straddle batches (S3 % TILE_VOX == 0).
            long long i0 = tile * (long long)TILE_VOX;        // voxel base
            int       b  = (i0 >= S3) ? 1 : 0;
            const float* fg = x + (long long)(2 * b + 1) * S3 // x[b,1]
                                + (i0 - (long long)b * S3) + (long long)tid * VPT;
            if (lab0.x | lab0.y | lab0.z | lab0.w) {
                const float4 f = *reinterpret_cast<const float4*>(fg);
                if (lab0.x > 0) { int c = lab0.x & (MAXC - 1);
                    atomicAdd(&s_tp[c], f.x); atomicAdd(&s_cnt[c], 1u); }
                if (lab0.y > 0) { int c = lab0.y & (MAXC - 1);
                    atomicAdd(&s_tp[c], f.y); atomicAdd(&s_cnt[c], 1u); }
                if (lab0.z > 0) { int c = lab0.z & (MAXC - 1);
                    atomicAdd(&s_tp[c], f.z); atomicAdd(&s_cnt[c], 1u); }
                if (lab0.w > 0) { int c = lab0.w & (MAXC - 1);
                    atomicAdd(&s_tp[c], f.w); atomicAdd(&s_cnt[c], 1u); }
            }
            if (lab1.x | lab1.y | lab1.z | lab1.w) {
                const float4 f = *reinterpret_cast<const float4*>(fg + 4);
                if (lab1.x > 0) { int c = lab1.x & (MAXC - 1);
                    atomicAdd(&s_tp[c], f.x); atomicAdd(&s_cnt[c], 1u); }
                if (lab1.y > 0) { int c = lab1.y & (MAXC - 1);
                    atomicAdd(&s_tp[c], f.y); atomicAdd(&s_cnt[c], 1u); }
                if (lab1.z > 0) { int c = lab1.z & (MAXC - 1);
                    atomicAdd(&s_tp[c], f.z); atomicAdd(&s_cnt[c], 1u); }
                if (lab1.w > 0) { int c = lab1.w & (MAXC - 1);
                    atomicAdd(&s_tp[c], f.w); atomicAdd(&s_cnt[c], 1u); }
            }
        }
        buf ^= 1;
    }

    __syncthreads();
    if (s_cnt[tid] != 0u) {
        atomicAdd(&g_tp[tid],  s_tp[tid]);
        atomicAdd(&g_cnt[tid], s_cnt[tid]);
    }
}

// ---------------------------------------------------------------------------
// Kernel 3: recall = sum_c tp[c]/cnt[c];  out = -(recall+1)/(ncomp+1).
// ---------------------------------------------------------------------------
__global__ void NoduleRecall_finalize(const float* __restrict__ tp,
                                      const unsigned* __restrict__ cnt,
                                      const int* __restrict__ ncomp_p,
                                      float* __restrict__ out) {
    __shared__ float red[MAXC];
    const int tid   = threadIdx.x;
    const int ncomp = *ncomp_p;

    float v = 0.0f;
    if (tid >= 1 && tid <= ncomp) {
        unsigned c = cnt[tid];
        if (c != 0u) v = tp[tid] / (float)c;
    }
    red[tid] = v;
    __syncthreads();
    for (int s = MAXC / 2; s > 0; s >>= 1) {
        if (tid < s) red[tid] += red[tid + s];
        __syncthreads();
    }
    if (tid == 0)
        out[0] = -(red[0] + 1.0f) / ((float)ncomp + 1.0f);
}

// ---------------------------------------------------------------------------
extern "C" void kernel_launch(void* const* d_in, const int* in_sizes, int n_in,
                              void* d_out, int out_size, void* d_ws, size_t ws_size,
                              hipStream_t stream) {
    const float* x      = (const float*)d_in[0];   // (B,2,S,S,S) f32
    /* d_in[1] = y : redundant with comp_labels>0, not read */
    const int*   labels = (const int*)d_in[2];     // (B,S,S,S) i32
    const int*   ncomp  = (const int*)d_in[3];     // scalar int

    float*    tp  = (float*)d_ws;
    unsigned* cnt = (unsigned*)((char*)d_ws + MAXC * sizeof(float));
    float*    out = (float*)d_out;

    const long long nvox   = 2LL * S3;             // == in_sizes[2]
    const long long ntiles = nvox / TILE_VOX;      // 6912, exact

    NoduleRecall_zero_ws<<<1, MAXC, 0, stream>>>(tp, cnt);

    // 6912 tiles / 1728 blocks = exactly 4 tiles per block.
    int grid = (int)((ntiles < 1728) ? ntiles : 1728);
    NoduleRecall_accum<<<grid, BLOCK, 0, stream>>>(labels, x, tp, cnt, ntiles);

    NoduleRecall_finalize<<<1, MAXC, 0, stream>>>(tp, cnt, ncomp, out);
}